// YoloNMS_75806172774675
// MI455X (gfx1250) — compile-verified
//
#include <hip/hip_runtime.h>

#define NANCH   102000
#define NCH     85
#define NCLS    80
#define KSEL    8192
#define TILES   512        // KSEL/16 tiles per matrix row
#define WPR16   512        // ushort halfwords per mask row
#define WPR32   256        // uint32 words per mask row
#define MAXDET  300
#define CONF_T  0.25f
#define IOU_T   0.45f

typedef __attribute__((ext_vector_type(2))) float v2f;
typedef __attribute__((ext_vector_type(8))) float v8f;

#if __has_builtin(__builtin_amdgcn_wmma_f32_16x16x4_f32)
#define HAVE_WMMA_F32X4 1
#else
#define HAVE_WMMA_F32X4 0
#endif

__device__ __forceinline__ unsigned ballot32(bool p) {
#if __has_builtin(__builtin_amdgcn_ballot_w32)
    return __builtin_amdgcn_ballot_w32(p);
#else
    return (unsigned)__ballot(p);
#endif
}

// ---------------------------------------------------------------------------
// Stage 1: per-anchor scores, argmax class, xywh -> yxyx boxes
// ---------------------------------------------------------------------------
__global__ void k_score(const float* __restrict__ p, float* __restrict__ scores,
                        float* __restrict__ classes, float4* __restrict__ boxes) {
    int i = blockIdx.x * 256 + threadIdx.x;
    if (i >= NANCH) return;
    const float* r = p + (long)i * NCH;
    // stream-ahead prefetch (global_prefetch_b8)
    __builtin_prefetch(r + (long)NCH * 256, 0, 1);
    float x = r[0], y = r[1], w = r[2], h = r[3], obj = r[4];
    float best = r[5];
    int bc = 0;
#pragma unroll 8
    for (int c = 1; c < NCLS; ++c) {
        float v = r[5 + c];
        if (v > best) { best = v; bc = c; }
    }
    scores[i]  = (obj > CONF_T) ? best * obj : 0.0f;
    classes[i] = (float)bc;
    float4 b;
    b.x = y - h * 0.5f;   // top y
    b.y = x - w * 0.5f;   // top x
    b.z = y + h * 0.5f;   // bottom y
    b.w = x + w * 0.5f;   // bottom x
    boxes[i] = b;
}

// ---------------------------------------------------------------------------
// Stage 2: exact top-K via two-level 16-bit radix select (scores >= 0)
// ---------------------------------------------------------------------------
__global__ void k_hist_hi(const float* __restrict__ scores, unsigned* __restrict__ hist) {
    int i = blockIdx.x * 256 + threadIdx.x;
    if (i >= NANCH) return;
    unsigned b = __float_as_uint(scores[i]) >> 16;
    atomicAdd(&hist[b], 1u);
}

__global__ void k_scan_hi(const unsigned* __restrict__ hist, unsigned* __restrict__ meta) {
    __shared__ unsigned psum[1024];
    int t = threadIdx.x;
    int hiBin = 65535 - t * 64;
    unsigned s = 0;
    for (int k = 0; k < 64; ++k) s += hist[hiBin - k];
    psum[t] = s;
    __syncthreads();
    if (t == 0) {
        unsigned cum = 0;
        int seg = 0;
        for (; seg < 1024; ++seg) {
            if (cum + psum[seg] >= (unsigned)KSEL) break;
            cum += psum[seg];
        }
        int hb = 65535 - seg * 64;
        int bin = hb;
        for (int k = 0; k < 64; ++k) {
            unsigned c = hist[hb - k];
            if (cum + c >= (unsigned)KSEL) { bin = hb - k; break; }
            cum += c;
        }
        meta[0] = (unsigned)bin;  // selected hi-16 bin
        meta[1] = cum;            // count strictly above that bin
    }
}

__global__ void k_hist_lo(const float* __restrict__ scores, unsigned* __restrict__ hist,
                          const unsigned* __restrict__ meta) {
    int i = blockIdx.x * 256 + threadIdx.x;
    if (i >= NANCH) return;
    unsigned b = __float_as_uint(scores[i]);
    if ((b >> 16) == meta[0]) atomicAdd(&hist[b & 0xFFFFu], 1u);
}

__global__ void k_scan_lo(const unsigned* __restrict__ hist, unsigned* __restrict__ meta) {
    __shared__ unsigned psum[1024];
    int t = threadIdx.x;
    int hiBin = 65535 - t * 64;
    unsigned s = 0;
    for (int k = 0; k < 64; ++k) s += hist[hiBin - k];
    psum[t] = s;
    __syncthreads();
    if (t == 0) {
        unsigned target = (unsigned)KSEL - meta[1];
        unsigned cum = 0;
        int seg = 0;
        for (; seg < 1024; ++seg) {
            if (cum + psum[seg] >= target) break;
            cum += psum[seg];
        }
        int hb = 65535 - seg * 64;
        int bin = hb;
        for (int k = 0; k < 64; ++k) {
            unsigned c = hist[hb - k];
            if (cum + c >= target) { bin = hb - k; break; }
            cum += c;
        }
        meta[2] = (meta[0] << 16) | (unsigned)bin;  // exact 32-bit threshold T
        meta[3] = target - cum;                     // number of == T entries to take
        meta[4] = 0;                                // n_sel counter
        meta[5] = 0;                                // n_eq counter
    }
}

__global__ void k_compact(const float* __restrict__ scores,
                          unsigned long long* __restrict__ selkeys,
                          unsigned* __restrict__ meta) {
    int i = blockIdx.x * 256 + threadIdx.x;
    if (i >= NANCH) return;
    unsigned b = __float_as_uint(scores[i]);
    unsigned T = meta[2];
    if (b > T) {
        unsigned slot = atomicAdd(&meta[4], 1u);
        selkeys[slot] = ((unsigned long long)b << 32) |
                        (unsigned long long)(0xFFFFFFFFu - (unsigned)i);
    } else if (b == T) {
        unsigned e = atomicAdd(&meta[5], 1u);
        if (e < meta[3]) {
            unsigned slot = atomicAdd(&meta[4], 1u);
            selkeys[slot] = ((unsigned long long)b << 32) |
                            (unsigned long long)(0xFFFFFFFFu - (unsigned)i);
        }
    }
}

// ---------------------------------------------------------------------------
// Stage 3: bitonic sort of 8192 64-bit keys fully in LDS (64 KB), then gather.
// Key = score_bits<<32 | ~idx  => descending sort gives score desc, idx asc.
// ---------------------------------------------------------------------------
__global__ void k_sort(const unsigned long long* __restrict__ selkeys,
                       const float* __restrict__ classes,
                       const float4* __restrict__ boxes,
                       float* __restrict__ sscores, float* __restrict__ scls,
                       float4* __restrict__ sboxes) {
    __shared__ unsigned long long keys[KSEL];
    int t = threadIdx.x;
#pragma unroll
    for (int e = 0; e < 8; ++e) keys[t + e * 1024] = selkeys[t + e * 1024];
    __syncthreads();
    for (int k = 2; k <= KSEL; k <<= 1) {
        for (int j = k >> 1; j > 0; j >>= 1) {
#pragma unroll
            for (int e = 0; e < 8; ++e) {
                int i = t + e * 1024;
                int ixj = i ^ j;
                if (ixj > i) {
                    unsigned long long a = keys[i], b = keys[ixj];
                    bool sw = ((i & k) == 0) ? (a < b) : (a > b);  // descending
                    if (sw) { keys[i] = b; keys[ixj] = a; }
                }
            }
            __syncthreads();
        }
    }
#pragma unroll
    for (int e = 0; e < 8; ++e) {
        int s = t + e * 1024;
        unsigned long long key = keys[s];
        unsigned sb = (unsigned)(key >> 32);
        unsigned idx = 0xFFFFFFFFu - (unsigned)(key & 0xFFFFFFFFu);
        sscores[s] = __uint_as_float(sb);
        scls[s]    = classes[idx];
        sboxes[s]  = boxes[idx];
    }
}

// ---------------------------------------------------------------------------
// Stage 4: IoU suppression bitmask. One wave per 16x16 tile.
// Union term area_i + area_j computed with v_wmma_f32_16x16x4_f32:
//   A (16x4 f32):  col0 = area_i, col1 = 1, cols 2..3 = 0
//   B (4x16 f32):  row0 = 1, row1 = area_j, rows 2..3 = 0
//   D[m][n] = area_i[m] + area_j[n]   (exact f32, RNE)
// D layout: lane = hi*16 + n, VGPR r => element (m = r + 8*hi, n).
// IoU threshold test done division-free: inter > IOU_T*(union + 1e-9).
// Row halfwords assembled via wave32 ballot, stored without atomics.
// ---------------------------------------------------------------------------
__global__ void k_iou(const float4* __restrict__ sboxes, unsigned short* __restrict__ mask) {
    __shared__ float4 rbox[8][16];
    __shared__ float4 cbox[8][16];
    int wv   = threadIdx.x >> 5;
    int lane = threadIdx.x & 31;
    int tile = blockIdx.x * 8 + wv;
    int tI = tile >> 9;          // / TILES
    int tJ = tile & (TILES - 1);
    int I0 = tI * 16, J0 = tJ * 16;

    if (lane < 16) rbox[wv][lane]      = sboxes[I0 + lane];
    else           cbox[wv][lane - 16] = sboxes[J0 + lane - 16];
    __syncthreads();

    int N  = lane & 15;
    int hi = lane >> 4;
    unsigned sel = 0u;

    if (J0 + 15 > I0) {  // tiles fully at/below the diagonal contribute nothing
        float areaR = 0.0f, areaC = 0.0f;
        if (lane < 16) {
            float4 bi = rbox[wv][lane];
            areaR = (bi.z - bi.x) * (bi.w - bi.y);
            float4 bj = cbox[wv][lane];
            areaC = (bj.z - bj.x) * (bj.w - bj.y);
        }
#if HAVE_WMMA_F32X4
        v2f A; A.x = areaR;                     A.y = (lane < 16) ? 1.0f : 0.0f;
        v2f B; B.x = (lane < 16) ? 1.0f : 0.0f; B.y = areaC;
        v8f C = {};
        v8f S = __builtin_amdgcn_wmma_f32_16x16x4_f32(false, A, false, B,
                                                      (short)0, C, false, false);
#endif
        float4 bj = cbox[wv][N];
#if !HAVE_WMMA_F32X4
        float aj = (bj.z - bj.x) * (bj.w - bj.y);
#endif
#pragma unroll
        for (int r = 0; r < 8; ++r) {
            int M = r + hi * 8;
            float4 bi = rbox[wv][M];
            float ty = fmaxf(bi.x, bj.x);
            float tx = fmaxf(bi.y, bj.y);
            float by = fminf(bi.z, bj.z);
            float bx = fminf(bi.w, bj.w);
            float inter = fmaxf(by - ty, 0.0f) * fmaxf(bx - tx, 0.0f);
#if HAVE_WMMA_F32X4
            float sum = S[r];
#else
            float sum = (bi.z - bi.x) * (bi.w - bi.y) + aj;
#endif
            // division-free threshold: inter/(union+eps) > IOU_T
            bool pred = (inter > IOU_T * ((sum - inter) + 1e-9f)) &&
                        ((J0 + N) > (I0 + M));
            unsigned bl = ballot32(pred);           // wave-uniform
            if ((lane & 7) == r) sel = bl;
        }
    }
    if (lane < 16) {
        unsigned hw = (lane < 8) ? (sel & 0xFFFFu) : (sel >> 16);
        mask[(size_t)(I0 + lane) * WPR16 + tJ] = (unsigned short)hw;
    }
}

// ---------------------------------------------------------------------------
// Stage 5: sequential greedy suppression scan (single block, LDS bitset)
// ---------------------------------------------------------------------------
__global__ void k_nms(const unsigned* __restrict__ mask, const float* __restrict__ sscores,
                      unsigned char* __restrict__ kept) {
    __shared__ unsigned removed[WPR32];
    __shared__ int flag;
    int t = threadIdx.x;
    removed[t] = 0u;
    __syncthreads();
    for (int i = 0; i < KSEL; ++i) {
        if (t == 0)
            flag = (((removed[i >> 5] >> (i & 31)) & 1u) == 0u) && (sscores[i] > 0.0f);
        __syncthreads();
        if (flag) removed[t] |= mask[(size_t)i * WPR32 + t];
        __syncthreads();
    }
    unsigned rw = removed[t];
#pragma unroll
    for (int b = 0; b < 32; ++b) {
        int j = t * 32 + b;
        kept[j] = ((((rw >> b) & 1u) == 0u) && (sscores[j] > 0.0f)) ? 1 : 0;
    }
}

// ---------------------------------------------------------------------------
// Stage 6: emit first 300 kept entries (input already score-descending)
// Output layout: boxes[300*4] | classes[300] | scores[300]
// ---------------------------------------------------------------------------
__global__ void k_final(const unsigned char* __restrict__ kept,
                        const float4* __restrict__ sboxes,
                        const float* __restrict__ scls,
                        const float* __restrict__ sscores,
                        float* __restrict__ out) {
    __shared__ unsigned cnt[256];
    int t = threadIdx.x;
    for (int i = t; i < 1800; i += 256) out[i] = 0.0f;
    unsigned c = 0;
    for (int e = 0; e < 32; ++e) c += kept[t * 32 + e];
    cnt[t] = c;
    __syncthreads();
    if (t == 0) {
        unsigned run = 0;
        for (int i = 0; i < 256; ++i) { unsigned v = cnt[i]; cnt[i] = run; run += v; }
    }
    __syncthreads();
    unsigned rank = cnt[t];
    for (int e = 0; e < 32; ++e) {
        int j = t * 32 + e;
        if (kept[j]) {
            if (rank < MAXDET) {
                float4 b = sboxes[j];
                out[rank * 4 + 0] = b.x;
                out[rank * 4 + 1] = b.y;
                out[rank * 4 + 2] = b.z;
                out[rank * 4 + 3] = b.w;
                out[1200 + rank] = scls[j];
                out[1500 + rank] = sscores[j];
            }
            ++rank;
        }
    }
}

// ---------------------------------------------------------------------------
extern "C" void kernel_launch(void* const* d_in, const int* in_sizes, int n_in,
                              void* d_out, int out_size, void* d_ws, size_t ws_size,
                              hipStream_t stream) {
    (void)in_sizes; (void)n_in; (void)out_size; (void)ws_size;
    const float* preds = (const float*)d_in[0];
    float* out = (float*)d_out;

    char* ws = (char*)d_ws;
    size_t off = 0;
    auto alloc = [&](size_t bytes) -> char* {
        char* p = ws + off;
        off += (bytes + 255) & ~(size_t)255;
        return p;
    };
    float*              scores  = (float*)alloc((size_t)NANCH * 4);
    float*              classes = (float*)alloc((size_t)NANCH * 4);
    float4*             boxes   = (float4*)alloc((size_t)NANCH * 16);
    unsigned*           hist    = (unsigned*)alloc((size_t)65536 * 4);
    unsigned*           meta    = (unsigned*)alloc(256);
    unsigned long long* selkeys = (unsigned long long*)alloc((size_t)KSEL * 8);
    float*              sscores = (float*)alloc((size_t)KSEL * 4);
    float*              scls    = (float*)alloc((size_t)KSEL * 4);
    float4*             sboxes  = (float4*)alloc((size_t)KSEL * 16);
    unsigned short*     mask    = (unsigned short*)alloc((size_t)KSEL * WPR16 * 2);
    unsigned char*      kept    = (unsigned char*)alloc((size_t)KSEL);

    int ablocks = (NANCH + 255) / 256;

    k_score<<<ablocks, 256, 0, stream>>>(preds, scores, classes, boxes);

    hipMemsetAsync(hist, 0, (size_t)65536 * 4, stream);
    k_hist_hi<<<ablocks, 256, 0, stream>>>(scores, hist);
    k_scan_hi<<<1, 1024, 0, stream>>>(hist, meta);
    hipMemsetAsync(hist, 0, (size_t)65536 * 4, stream);
    k_hist_lo<<<ablocks, 256, 0, stream>>>(scores, hist, meta);
    k_scan_lo<<<1, 1024, 0, stream>>>(hist, meta);
    k_compact<<<ablocks, 256, 0, stream>>>(scores, selkeys, meta);

    k_sort<<<1, 1024, 0, stream>>>(selkeys, classes, boxes, sscores, scls, sboxes);

    k_iou<<<(TILES * TILES) / 8, 256, 0, stream>>>(sboxes, mask);
    k_nms<<<1, 256, 0, stream>>>((const unsigned*)mask, sscores, kept);
    k_final<<<1, 256, 0, stream>>>(kept, sboxes, scls, sscores, out);
}